// New_DistMaps_54924041781818
// MI455X (gfx1250) — compile-verified
//
#include <hip/hip_runtime.h>
#include <hip/hip_bf16.h>
#include <stdint.h>

#define BB   8
#define PP   24
#define HH   448
#define WW   448
#define NCLS 19

// ---------------------------------------------------------------------------
// Kernel 1: zero-fill the (B, NCLS, H, W) canvas at HBM rate.
// Uses CDNA5 async LDS->global 128-bit stores (tracked by ASYNCcnt): data is
// sourced from a single zeroed 16-byte LDS slot (broadcast read by all 32
// lanes), so the store stream carries no VGPR data operand. Drained with
// s_wait_asynccnt 0 before wave exit; the next kernel on the stream is
// ordered after dispatch completion.
// ---------------------------------------------------------------------------
__global__ __launch_bounds__(256) void dm_fill_zero_async(float* __restrict__ out,
                                                          long long n4,
                                                          long long n_total) {
    __shared__ float4 zslot;
    if (threadIdx.x == 0) {
        zslot = make_float4(0.0f, 0.0f, 0.0f, 0.0f);
    }
    __syncthreads();

    // Low 32 bits of a generic pointer to LDS == LDS byte offset (aperture in
    // the high bits), per the gfx1250 flat-address aperture rules.
    uint32_t lds_off = (uint32_t)(uintptr_t)&zslot;

    long long i      = (long long)blockIdx.x * blockDim.x + threadIdx.x;
    long long stride = (long long)gridDim.x * blockDim.x;
    const float4* base = (const float4*)out;

    for (; i < n4; i += stride) {
        uint64_t addr = (uint64_t)(uintptr_t)(base + i);
        asm volatile("global_store_async_from_lds_b128 %0, %1, off"
                     :
                     : "v"(addr), "v"(lds_off)
                     : "memory");
    }

    // Scalar tail (n_total not divisible by 4) — handled by block 0 lane set.
    long long tail_start = n4 * 4;
    for (long long t = tail_start + (long long)blockIdx.x * blockDim.x + threadIdx.x;
         t < n_total;
         t += stride) {
        out[t] = 0.0f;
    }

    asm volatile("s_wait_asynccnt 0" ::: "memory");
}

// ---------------------------------------------------------------------------
// Kernel 2: paint 255.0 disks. One block per (batch, point); 128 threads
// cover the 11x11 bounding box of the radius-5 disk. Coordinates are
// integral-valued floats, so enumerating rows/cols at pr+dy, pc+dx with
// integer dy,dx in [-5,5] and testing dy^2+dx^2 <= 25 reproduces the
// reference mask exactly. Overlapping writes all store 255.0f -> race-safe.
// ---------------------------------------------------------------------------
__global__ __launch_bounds__(128) void dm_paint_disks(const float* __restrict__ coords,
                                                      float* __restrict__ out) {
    int bp = blockIdx.x;              // 0 .. BB*PP-1
    int b  = bp / PP;

    const float* c3 = coords + (size_t)bp * 3;
    float pr = c3[0];
    float pc = c3[1];
    if (fmaxf(pr, pc) < 0.0f) return;          // invalid point

    int cls = (int)c3[2];                       // 0..18 by construction

    int t = threadIdx.x;
    if (t >= 121) return;
    int dy = t / 11 - 5;
    int dx = t % 11 - 5;
    if (dy * dy + dx * dx > 25) return;         // outside disk (<= 25 kept)

    int r = __float2int_rn(pr) + dy;
    int q = __float2int_rn(pc) + dx;
    if (r < 0 || r >= HH || q < 0 || q >= WW) return;

    size_t idx = (((size_t)b * NCLS + (size_t)cls) * HH + (size_t)r) * WW + (size_t)q;
    out[idx] = 255.0f;
}

extern "C" void kernel_launch(void* const* d_in, const int* in_sizes, int n_in,
                              void* d_out, int out_size, void* d_ws, size_t ws_size,
                              hipStream_t stream) {
    (void)d_ws; (void)ws_size; (void)n_in; (void)in_sizes;

    // d_in[0] = x (unused except for shape), d_in[1] = coords (B,P,3) float32
    const float* coords = (const float*)d_in[1];
    float* out = (float*)d_out;

    long long n_total = (long long)out_size;     // 8*19*448*448 = 30,507,008
    long long n4      = n_total >> 2;            // float4 count

    dm_fill_zero_async<<<2048, 256, 0, stream>>>(out, n4, n_total);
    dm_paint_disks<<<BB * PP, 128, 0, stream>>>(coords, out);
}